// NodeModel_45492293599374
// MI455X (gfx1250) — compile-verified
//
#include <hip/hip_runtime.h>

// GNN NodeModel: scatter(max/mean/add) + concat + [Linear+BN+ReLU] + [Linear+BN]
// fp32 end-to-end; GEMMs on V_WMMA_F32_16X16X4_F32 (native fp32 matrix path).

#define N_NODES 50000
#define N_EDGES 800000
#define X_IND   512
#define EDGE_IND 64
#define U_IND   128
#define HS      1024
#define CIN     832      // 512 + 128 + 3*64
#define BN_EPS  1e-5f

// h column layout: [0,512) x | [512,640) u[batch] | [640,704) smax | [704,768) smean | [768,832) ssum
#define OFF_U    512
#define OFF_MAX  640
#define OFF_MEAN 704
#define OFF_SUM  768

typedef float v2f __attribute__((ext_vector_type(2)));
typedef float v8f __attribute__((ext_vector_type(8)));

// ---------------- graph aggregation ----------------

__global__ void init_h_kernel(const float* __restrict__ x, const float* __restrict__ u,
                              const int* __restrict__ batch, float* __restrict__ h,
                              float* __restrict__ cnt) {
  int n = blockIdx.x;
  int g = batch[n];
  float* hr = h + (size_t)n * CIN;
  const float* xr = x + (size_t)n * X_IND;
  const float* ur = u + (size_t)g * U_IND;
  for (int c = threadIdx.x; c < CIN; c += blockDim.x) {
    float v;
    if (c < OFF_U)        v = xr[c];
    else if (c < OFF_MAX) v = ur[c - OFF_U];
    else if (c < OFF_MEAN) v = -__builtin_inff();  // smax identity
    else                  v = 0.0f;                // smean/ssum
    hr[c] = v;
  }
  if (threadIdx.x == 0) cnt[n] = 0.0f;
}

// order-preserving float max via signed/unsigned int atomics (always lowers natively)
__device__ inline void atomicMaxFloat(float* addr, float v) {
  if (v >= 0.0f) atomicMax((int*)addr, __float_as_int(v));
  else           atomicMin((unsigned int*)addr, __float_as_uint(v));
}

__global__ void scatter_kernel(const float* __restrict__ ea, const int* __restrict__ ei,
                               float* __restrict__ h, float* __restrict__ cnt) {
  int t = blockIdx.x * blockDim.x + threadIdx.x;
  int e = t >> 6, f = t & 63;
  if (e >= N_EDGES) return;
  int col = ei[N_EDGES + e];                    // edge_index[1][e]
  float v = ea[(size_t)e * EDGE_IND + f];
  float* hr = h + (size_t)col * CIN;
  atomicAdd(hr + OFF_SUM + f, v);               // global_atomic_add_f32
  atomicMaxFloat(hr + OFF_MAX + f, v);
  if (f == 0) atomicAdd(cnt + col, 1.0f);
}

__global__ void finalize_kernel(const float* __restrict__ cnt, float* __restrict__ h) {
  int t = blockIdx.x * blockDim.x + threadIdx.x;
  int n = t >> 6, f = t & 63;
  if (n >= N_NODES) return;
  float c = cnt[n];
  float* hr = h + (size_t)n * CIN;
  float mx = hr[OFF_MAX + f];
  hr[OFF_MAX + f] = (c > 0.0f) ? mx : 0.0f;     // empty segments -> 0 (torch_scatter)
  hr[OFF_MEAN + f] = hr[OFF_SUM + f] / fmaxf(c, 1.0f);
}

// ---------------- fp32 WMMA GEMM:  C[M,1024] = A[M,K] @ W[1024,K]^T + bias ----------------
// 128x128 block tile, 8 wave32s; each wave owns a 16-row strip x 128 cols
// (8 accumulators of v8f).  K staged 32 at a time through LDS.

#define MT 128
#define NT 128
#define KT 32
#define LDA 36   // padded stride (floats): 144B rows -> 16B-aligned float4 stores, bank spread

__global__ __launch_bounds__(256) void gemm_wmma_f32(
    const float* __restrict__ A, const float* __restrict__ W,
    const float* __restrict__ bias, float* __restrict__ C, int M, int K) {
  __shared__ float As[MT * LDA];
  __shared__ float Bs[NT * LDA];

  const int tid  = threadIdx.x;
  const int wave = tid >> 5;
  const int lane = tid & 31;
  const int half = lane >> 4;     // K-pair select (A/B frags), M+8 select (C/D)
  const int mr   = lane & 15;     // row (A) / col (B,C,D) within 16

  const long mBase = (long)blockIdx.x * MT;
  const int  nBase = blockIdx.y * NT;

  v8f acc[8];
#pragma unroll
  for (int i = 0; i < 8; ++i)
#pragma unroll
    for (int j = 0; j < 8; ++j) acc[i][j] = 0.0f;

  const int stages = K / KT;      // K is 832 or 1024, both % 32 == 0
  for (int s = 0; s < stages; ++s) {
    const int k0 = s * KT;
    // stage A tile (guard M) and B tile (HS rows, never OOB) into LDS, float4 each
#pragma unroll
    for (int i = tid; i < MT * (KT / 4); i += 256) {
      int r = i >> 3, c4 = (i & 7) << 2;
      long row = mBase + r;
      float4 v = make_float4(0.f, 0.f, 0.f, 0.f);
      if (row < M) v = *(const float4*)(A + row * (long)K + k0 + c4);
      *(float4*)(&As[r * LDA + c4]) = v;
      float4 w = *(const float4*)(W + (long)(nBase + r) * K + k0 + c4);
      *(float4*)(&Bs[r * LDA + c4]) = w;
    }
    __syncthreads();

#pragma unroll
    for (int k = 0; k < KT; k += 4) {
      const int ko = k + half * 2;
      v2f a = *(const v2f*)&As[(wave * 16 + mr) * LDA + ko];   // A 16x4: lane=M, half=K-pair
#pragma unroll
      for (int nt = 0; nt < 8; ++nt) {
        v2f b = *(const v2f*)&Bs[(nt * 16 + mr) * LDA + ko];   // B 4x16: lane=N, half=K-pair
        acc[nt] = __builtin_amdgcn_wmma_f32_16x16x4_f32(
            false, a, false, b, (short)0, acc[nt], false, false);
      }
    }
    __syncthreads();
  }

  // epilogue: C/D layout -> row = v + 8*half, col = mr; fuse bias
#pragma unroll
  for (int nt = 0; nt < 8; ++nt) {
    const int col = nBase + nt * 16 + mr;
    const float bv = bias[col];
#pragma unroll
    for (int v = 0; v < 8; ++v) {
      long row = mBase + wave * 16 + v + half * 8;
      if (row < M) C[row * (long)HS + col] = acc[nt][v] + bv;
    }
  }
}

// ---------------- BatchNorm (training stats over N rows, per channel) ----------------

__global__ void zero_stats_kernel(float* __restrict__ stats) {
  for (int i = threadIdx.x; i < 2 * HS; i += blockDim.x) stats[i] = 0.0f;
}

__global__ void bn_stats_kernel(const float* __restrict__ y, float* __restrict__ stats, int M) {
  const int c4 = threadIdx.x * 4;                 // 256 threads * 4 = 1024 channels
  const long r0 = (long)blockIdx.x * 256;
  const long r1 = (r0 + 256 < M) ? r0 + 256 : M;
  float4 s = make_float4(0.f, 0.f, 0.f, 0.f);
  float4 q = make_float4(0.f, 0.f, 0.f, 0.f);
  for (long r = r0; r < r1; ++r) {
    float4 v = *(const float4*)(y + r * HS + c4);
    s.x += v.x; s.y += v.y; s.z += v.z; s.w += v.w;
    q.x += v.x * v.x; q.y += v.y * v.y; q.z += v.z * v.z; q.w += v.w * v.w;
  }
  atomicAdd(&stats[c4 + 0], s.x); atomicAdd(&stats[c4 + 1], s.y);
  atomicAdd(&stats[c4 + 2], s.z); atomicAdd(&stats[c4 + 3], s.w);
  atomicAdd(&stats[HS + c4 + 0], q.x); atomicAdd(&stats[HS + c4 + 1], q.y);
  atomicAdd(&stats[HS + c4 + 2], q.z); atomicAdd(&stats[HS + c4 + 3], q.w);
}

__global__ void bn_apply_kernel(float* __restrict__ y, const float* __restrict__ stats,
                                const float* __restrict__ g, const float* __restrict__ be,
                                int M, int relu) {
  long idx = (long)blockIdx.x * blockDim.x + threadIdx.x;   // one float4 per thread
  if (idx >= (long)M * (HS / 4)) return;
  long r = idx >> 8;
  int c4 = (int)(idx & 255) * 4;
  const float invN = 1.0f / (float)M;
  float4 v  = *(float4*)(y + r * HS + c4);
  float4 sm = *(const float4*)(stats + c4);
  float4 sq = *(const float4*)(stats + HS + c4);
  float4 gg = *(const float4*)(g + c4);
  float4 bb = *(const float4*)(be + c4);
  float m, var, sc;
  m = sm.x * invN; var = sq.x * invN - m * m; sc = gg.x * __frsqrt_rn(var + BN_EPS);
  v.x = (v.x - m) * sc + bb.x;
  m = sm.y * invN; var = sq.y * invN - m * m; sc = gg.y * __frsqrt_rn(var + BN_EPS);
  v.y = (v.y - m) * sc + bb.y;
  m = sm.z * invN; var = sq.z * invN - m * m; sc = gg.z * __frsqrt_rn(var + BN_EPS);
  v.z = (v.z - m) * sc + bb.z;
  m = sm.w * invN; var = sq.w * invN - m * m; sc = gg.w * __frsqrt_rn(var + BN_EPS);
  v.w = (v.w - m) * sc + bb.w;
  if (relu) {
    v.x = fmaxf(v.x, 0.f); v.y = fmaxf(v.y, 0.f);
    v.z = fmaxf(v.z, 0.f); v.w = fmaxf(v.w, 0.f);
  }
  *(float4*)(y + r * HS + c4) = v;
}

// ---------------- launch ----------------

extern "C" void kernel_launch(void* const* d_in, const int* in_sizes, int n_in,
                              void* d_out, int out_size, void* d_ws, size_t ws_size,
                              hipStream_t stream) {
  const float* x   = (const float*)d_in[0];
  const float* ea  = (const float*)d_in[1];
  const float* u   = (const float*)d_in[2];
  const float* w1  = (const float*)d_in[3];
  const float* b1  = (const float*)d_in[4];
  const float* g1  = (const float*)d_in[5];
  const float* be1 = (const float*)d_in[6];
  const float* w2  = (const float*)d_in[7];
  const float* b2  = (const float*)d_in[8];
  const float* g2  = (const float*)d_in[9];
  const float* be2 = (const float*)d_in[10];
  const int*   ei  = (const int*)d_in[11];
  const int*   bat = (const int*)d_in[12];
  float* out = (float*)d_out;

  // workspace layout (bytes)
  char* ws = (char*)d_ws;
  const size_t off_h    = 0;                                        // 50000*832*4  = 166,400,000
  const size_t off_y1   = off_h  + (size_t)N_NODES * CIN * 4;       // 50000*1024*4 = 204,800,000
  const size_t off_cnt  = off_y1 + (size_t)N_NODES * HS * 4;        // 50000*4
  const size_t off_stat = off_cnt + ((size_t)N_NODES * 4 + 255 & ~(size_t)255);
  float* h     = (float*)(ws + off_h);
  float* y1    = (float*)(ws + off_y1);
  float* cnt   = (float*)(ws + off_cnt);
  float* stats = (float*)(ws + off_stat);                           // 2*1024 floats

  // 1) build h = [x | u[batch] | smax | smean | ssum]
  init_h_kernel<<<N_NODES, 256, 0, stream>>>(x, u, bat, h, cnt);
  scatter_kernel<<<(N_EDGES * 64) / 256, 256, 0, stream>>>(ea, ei, h, cnt);
  finalize_kernel<<<(N_NODES * 64 + 255) / 256, 256, 0, stream>>>(cnt, h);

  // 2) y1 = h @ w1^T + b1 ; BN(train) ; ReLU
  dim3 gemm_grid((N_NODES + MT - 1) / MT, HS / NT);
  gemm_wmma_f32<<<gemm_grid, 256, 0, stream>>>(h, w1, b1, y1, N_NODES, CIN);
  zero_stats_kernel<<<1, 256, 0, stream>>>(stats);
  bn_stats_kernel<<<(N_NODES + 255) / 256, 256, 0, stream>>>(y1, stats, N_NODES);
  bn_apply_kernel<<<(N_NODES * (HS / 4) + 255) / 256, 256, 0, stream>>>(
      y1, stats, g1, be1, N_NODES, 1);

  // 3) out = y1 @ w2^T + b2 ; BN(train)
  gemm_wmma_f32<<<gemm_grid, 256, 0, stream>>>(y1, w2, b2, out, N_NODES, HS);
  zero_stats_kernel<<<1, 256, 0, stream>>>(stats);
  bn_stats_kernel<<<(N_NODES + 255) / 256, 256, 0, stream>>>(out, stats, N_NODES);
  bn_apply_kernel<<<(N_NODES * (HS / 4) + 255) / 256, 256, 0, stream>>>(
      out, stats, g2, be2, N_NODES, 0);
}